// graph2linegraph_25348896981201
// MI455X (gfx1250) — compile-verified
//
#include <hip/hip_runtime.h>
#include <hip/hip_bf16.h>
#include <stdint.h>

#define N_NODES 10000
#define N_EDGES 25000
#define D_FEAT  128
#define TILE    2048
#define NTILES  ((N_EDGES + TILE - 1) / TILE)   // 13

__device__ __forceinline__ unsigned lds_off(const void* p) {
  // Generic pointers to LDS carry the byte offset in the low 32 bits.
  return (unsigned)(uintptr_t)p;
}

// Issue async global->LDS copies (gfx1250 GLOBAL_LOAD_ASYNC_TO_LDS_B128):
// n dwords (n % 4 == 0) from gsrc into dstLds, 16B per lane-issue, strided
// across all threads of the block. Tracked by ASYNCcnt; no VGPR data path.
__device__ __forceinline__ void async_issue(int* dstLds, const int* gsrc, int n) {
#if defined(__HIP_DEVICE_COMPILE__)
  int chunks = n >> 2;                              // 16-byte chunks
  for (int c = threadIdx.x; c < chunks; c += blockDim.x) {
    unsigned lds = lds_off(dstLds + c * 4);
    unsigned long long ga = (unsigned long long)(uintptr_t)(gsrc + c * 4);
    asm volatile("global_load_async_to_lds_b128 %0, %1, off"
                 :
                 : "v"(lds), "v"(ga)
                 : "memory");
  }
#else
  (void)dstLds; (void)gsrc; (void)n;
#endif
}

// Wait for this wave's outstanding async LDS copies.
__device__ __forceinline__ void async_wait0() {
#if defined(__HIP_DEVICE_COMPILE__)
  asm volatile("s_wait_asynccnt 0" ::: "memory");
#endif
}

// ---- Kernel 1: new_x = 0.25*(x[c0]+x[c1]) + 0.5*edge_attr -----------------
__global__ void k_newx(const float* __restrict__ x, const float* __restrict__ ea,
                       const int* __restrict__ ei, float* __restrict__ out) {
  int e = blockIdx.x * 8 + (threadIdx.x >> 5);    // one wave32 per edge row
  if (e >= N_EDGES) return;
  int lane = threadIdx.x & 31;
  int c0 = ei[e];
  int c1 = ei[N_EDGES + e];
  const float4* x0 = (const float4*)(x + (size_t)c0 * D_FEAT);
  const float4* x1 = (const float4*)(x + (size_t)c1 * D_FEAT);
  const float4* a  = (const float4*)(ea + (size_t)e * D_FEAT);
  float4* o        = (float4*)(out + (size_t)e * D_FEAT);
  float4 v0 = x0[lane], v1 = x1[lane], va = a[lane];
  float4 r;
  r.x = (v0.x + v1.x) * 0.25f + va.x * 0.5f;
  r.y = (v0.y + v1.y) * 0.25f + va.y * 0.5f;
  r.z = (v0.z + v1.z) * 0.25f + va.z * 0.5f;
  r.w = (v0.w + v1.w) * 0.25f + va.w * 0.5f;
  o[lane] = r;
}

// ---- Kernel 2: per-row match counts.  One wave32 per row; j split across
// lanes; count = sum of popc(ballot).  Tiles staged in LDS with async
// global->LDS copies, double-buffered: tile t+1 is issued before computing
// on tile t and waited only at the top of the next iteration. --------------
__global__ void k_count(const int* __restrict__ ei, int* __restrict__ counts) {
  __shared__ int tiles[2][2 * TILE];               // [buf][c0 | c1]
  int wave = threadIdx.x >> 5;
  int lane = threadIdx.x & 31;
  int i = blockIdx.x * 8 + wave;                   // one wave per row i
  int c0i = 0, c1i = -1;
  if (i < N_EDGES) { c0i = ei[i]; c1i = ei[N_EDGES + i]; }
  {
    int n0 = TILE < N_EDGES ? TILE : N_EDGES;
    async_issue(&tiles[0][0],    ei,           n0);
    async_issue(&tiles[0][TILE], ei + N_EDGES, n0);
  }
  int cnt = 0;
  for (int t = 0; t < NTILES; ++t) {
    int j0 = t * TILE;
    int n = min(TILE, N_EDGES - j0);
    async_wait0();                                 // this wave's copies for tile t done
    __syncthreads();                               // -> whole tile visible to block
    if (t + 1 < NTILES) {                          // prefetch tile t+1
      int j1 = j0 + TILE;
      int n1 = min(TILE, N_EDGES - j1);
      int b = (t + 1) & 1;
      async_issue(&tiles[b][0],    ei + j1,           n1);
      async_issue(&tiles[b][TILE], ei + N_EDGES + j1, n1);
    }
    const int* t0 = tiles[t & 1];
    const int* t1 = tiles[t & 1] + TILE;
    if (i < N_EDGES) {
      int nb = (n + 31) & ~31;                     // uniform bound, lanes converged
      for (int jj = lane; jj < nb; jj += 32) {
        bool p = (jj < n) && (c1i == t0[jj]) && (c0i != t1[jj]);
        cnt += __builtin_popcount(__builtin_amdgcn_ballot_w32(p));
      }
    }
    __syncthreads();                               // all reads of buf done
  }
  if (i < N_EDGES && lane == 0) counts[i] = cnt;
}

// ---- Kernel 3: single-block exclusive scan of 25000 counts ----------------
__global__ void k_scan(const int* __restrict__ counts, int* __restrict__ offsets,
                       int* __restrict__ totalp) {
  __shared__ int part[1024];
  const int CH = (N_EDGES + 1023) / 1024;          // 25 elements per thread
  int t = threadIdx.x;
  int b = t * CH;
  int s = 0;
  for (int k = 0; k < CH; ++k) {
    int idx = b + k;
    if (idx < N_EDGES) s += counts[idx];
  }
  part[t] = s;
  __syncthreads();
  for (int off = 1; off < 1024; off <<= 1) {       // Hillis-Steele inclusive scan
    int v = (t >= off) ? part[t - off] : 0;
    __syncthreads();
    part[t] += v;
    __syncthreads();
  }
  int run = part[t] - s;                           // exclusive prefix at chunk start
  for (int k = 0; k < CH; ++k) {
    int idx = b + k;
    if (idx < N_EDGES) { offsets[idx] = run; run += counts[idx]; }
  }
  if (t == 1023) { offsets[N_EDGES] = part[1023]; *totalp = part[1023]; }
}

// ---- Kernel 4: ordered fill of rows/cols (+attr index) via wave32 ballot.
// Same wave-per-row / double-buffered async-LDS structure as k_count. ------
__global__ void k_fill(const int* __restrict__ ei, const int* __restrict__ offsets,
                       float* __restrict__ out, int* __restrict__ attr_idx, int E) {
  __shared__ int tiles[2][2 * TILE];
  int wave = threadIdx.x >> 5;
  int lane = threadIdx.x & 31;
  int i = blockIdx.x * 8 + wave;                   // one wave per row i
  int c0i = 0, c1i = -1, base = 0;
  if (i < N_EDGES) { c0i = ei[i]; c1i = ei[N_EDGES + i]; base = offsets[i]; }
  float* rows_out = out + (size_t)N_EDGES * D_FEAT;
  float* cols_out = rows_out + E;
  unsigned lmask = (1u << lane) - 1u;
  {
    int n0 = TILE < N_EDGES ? TILE : N_EDGES;
    async_issue(&tiles[0][0],    ei,           n0);
    async_issue(&tiles[0][TILE], ei + N_EDGES, n0);
  }
  int cnt = 0;
  for (int t = 0; t < NTILES; ++t) {
    int j0 = t * TILE;
    int n = min(TILE, N_EDGES - j0);
    async_wait0();
    __syncthreads();
    if (t + 1 < NTILES) {
      int j1 = j0 + TILE;
      int n1 = min(TILE, N_EDGES - j1);
      int b = (t + 1) & 1;
      async_issue(&tiles[b][0],    ei + j1,           n1);
      async_issue(&tiles[b][TILE], ei + N_EDGES + j1, n1);
    }
    const int* t0 = tiles[t & 1];
    const int* t1 = tiles[t & 1] + TILE;
    if (i < N_EDGES) {
      int nb = (n + 31) & ~31;
      for (int jj = lane; jj < nb; jj += 32) {
        bool p = (jj < n) && (c1i == t0[jj]) && (c0i != t1[jj]);
        unsigned m = __builtin_amdgcn_ballot_w32(p);
        if (p) {
          int off = base + cnt + __builtin_popcount(m & lmask);
          rows_out[off] = (float)i;                // exact: values < 2^24
          cols_out[off] = (float)(j0 + jj);
          attr_idx[off] = c1i;                     // lg_edge_attr_idx = col1[row]
        }
        cnt += __builtin_popcount(m);
      }
    }
    __syncthreads();
  }
}

// ---- Kernel 5: lg_edge_attr[k] = new_x[attr_idx[k]] (float4 row copy) -----
__global__ void k_attr(const float* __restrict__ newx, const int* __restrict__ attr_idx,
                       float* __restrict__ attr_out, int E) {
  int k = blockIdx.x * 8 + (threadIdx.x >> 5);     // one wave32 per lg edge
  if (k >= E) return;
  int lane = threadIdx.x & 31;
  int idx = attr_idx[k];
  const float4* src = (const float4*)(newx + (size_t)idx * D_FEAT);
  float4* dst       = (float4*)(attr_out + (size_t)k * D_FEAT);
  dst[lane] = src[lane];
}

extern "C" void kernel_launch(void* const* d_in, const int* in_sizes, int n_in,
                              void* d_out, int out_size, void* d_ws, size_t ws_size,
                              hipStream_t stream) {
  const float* x  = (const float*)d_in[0];
  const float* ea = (const float*)d_in[1];
  const int*   ei = (const int*)d_in[2];           // [2, N_EDGES]: col0 then col1
  float* out = (float*)d_out;

  // Output layout: new_x (25000*128) | rows (E) | cols (E) | lg_edge_attr (E*128)
  int E = (out_size - N_EDGES * D_FEAT) / (D_FEAT + 2);
  if (E < 0) E = 0;

  int* counts   = (int*)d_ws;                      // [N_EDGES]
  int* offsets  = counts + N_EDGES;                // [N_EDGES+1]
  int* totalp   = offsets + N_EDGES + 1;           // [1]
  int* attr_idx = totalp + 1;                      // [E]

  k_newx <<<(N_EDGES + 7) / 8, 256, 0, stream>>>(x, ea, ei, out);
  k_count<<<(N_EDGES + 7) / 8, 256, 0, stream>>>(ei, counts);
  k_scan <<<1, 1024, 0, stream>>>(counts, offsets, totalp);
  k_fill <<<(N_EDGES + 7) / 8, 256, 0, stream>>>(ei, offsets, out, attr_idx, E);
  if (E > 0) {
    float* attr_out = out + (size_t)N_EDGES * D_FEAT + 2 * (size_t)E;
    k_attr<<<(E + 7) / 8, 256, 0, stream>>>(out, attr_idx, attr_out, E);
  }
}